// JPEG_Layer_66649302499797
// MI455X (gfx1250) — compile-verified
//
#include <hip/hip_runtime.h>

typedef __attribute__((ext_vector_type(2))) float v2f;
typedef __attribute__((ext_vector_type(8))) float v8f;

// 8-point DCT-II matrix, rows scaled: D[0][:] = 1/sqrt(8), D[i][j] = 0.5*cos(pi*(2j+1)*i/16)
#define H1 0.49039264020161522f
#define H2 0.46193976625564337f
#define H3 0.41573480615127262f
#define H4 0.35355339059327373f
#define H5 0.27778511650980114f
#define H6 0.19134171618254492f
#define H7 0.09754516100806417f

__device__ __constant__ float D8c[64] = {
    H4,  H4,  H4,  H4,  H4,  H4,  H4,  H4,
    H1,  H3,  H5,  H7, -H7, -H5, -H3, -H1,
    H2,  H6, -H6, -H2, -H2, -H6,  H6,  H2,
    H3, -H7, -H1, -H5,  H5,  H1,  H7, -H3,
    H4, -H4, -H4,  H4,  H4, -H4, -H4,  H4,
    H5, -H1,  H7,  H3, -H3, -H7,  H1, -H5,
    H6, -H2,  H2, -H6, -H6,  H2, -H2,  H6,
    H7, -H5,  H3, -H1,  H1, -H3,  H5, -H7
};

static __device__ __forceinline__ v8f wmma4(v2f a, v2f b, v8f c) {
    // D = A(16x4,f32) * B(4x16,f32) + C(16x16,f32)
    return __builtin_amdgcn_wmma_f32_16x16x4_f32(
        false, a, false, b, (short)0, c, false, false);
}

// Convert accumulator (C/D layout: vgpr=M row, lane=N col) into the B operand
// for K-chunk Q (rows 4Q..4Q+3).  Rows<8 live in lanes 0-15, rows>=8 in lanes
// 16-31, so conversion is a half-wave lane shift + select.
template <int Q>
static __device__ __forceinline__ v2f cd_to_b(v8f u, int lane) {
    v2f r;
    if constexpr (Q < 2) {
        float s0 = __shfl(u[4 * Q + 2], lane ^ 16, 32);
        float s1 = __shfl(u[4 * Q + 3], lane ^ 16, 32);
        r[0] = (lane < 16) ? u[4 * Q + 0] : s0;
        r[1] = (lane < 16) ? u[4 * Q + 1] : s1;
    } else {
        float s0 = __shfl(u[(4 * Q + 0) & 7], lane ^ 16, 32);
        float s1 = __shfl(u[(4 * Q + 1) & 7], lane ^ 16, 32);
        r[0] = (lane < 16) ? s0 : u[(4 * Q + 2) & 7];
        r[1] = (lane < 16) ? s1 : u[(4 * Q + 3) & 7];
    }
    return r;
}

#define LPAD 20  // LDS row pitch (floats); stride 20 -> conflict-free b64 reads

__global__ __launch_bounds__(256) void jpeg_wmma_kernel(
    const float* __restrict__ img, const float* __restrict__ Qy,
    const float* __restrict__ Qc, float* __restrict__ out, int nImgs) {
    __shared__ __align__(16) float lds[8][16 * LPAD];

    const int lane = threadIdx.x & 31;
    const int wid  = threadIdx.x >> 5;

    const int total = nImgs * 64 * 64;
    const int tile  = blockIdx.x * 8 + wid;
    if (tile >= total) return;  // wave-uniform, no block barriers used

    const int tx = tile & 63;
    const int t2 = tile >> 6;
    const int ty = t2 & 63;
    const int bi = t2 >> 6;

    const int m      = lane & 15;        // A-layout: M row held by this lane
    const int koff   = (lane >> 4) * 2;  // A-layout: K sub-offset per half-wave
    const int colN   = lane & 15;        // C/D layout: N col held by this lane
    const int rowOff = (lane >> 4) * 8;  // C/D layout: M base per half-wave

    // Per-lane fragments of DD = diag(D, D):
    //   dd[q]  : A-layout of DD       == B-layout of DD^T
    //   ddt[q] : A-layout of DD^T     == B-layout of DD
    v2f dd[4], ddt[4];
#pragma unroll
    for (int q = 0; q < 4; ++q) {
#pragma unroll
        for (int j = 0; j < 2; ++j) {
            int k = 4 * q + koff + j;
            float fdd = 0.f, fddt = 0.f;
            if (((m ^ k) & 8) == 0) {  // block-diagonal
                fdd  = D8c[(m & 7) * 8 + (k & 7)];
                fddt = D8c[(k & 7) * 8 + (m & 7)];
            }
            dd[q][j]  = fdd;
            ddt[q][j] = fddt;
        }
    }

    // ---- Load 16x16 tile, 3 channels, directly in WMMA A layout ----
    const size_t plane = 1024u * 1024u;
    const size_t base0 =
        (size_t)bi * 3 * plane + (size_t)(ty * 16 + m) * 1024 + (size_t)tx * 16;
    float ch[3][8];
#pragma unroll
    for (int c = 0; c < 3; ++c) {
        const float* p = img + base0 + (size_t)c * plane;
#pragma unroll
        for (int q = 0; q < 4; ++q) {
            v2f t = *reinterpret_cast<const v2f*>(p + 4 * q + koff);
            ch[c][2 * q + 0] = t[0];
            ch[c][2 * q + 1] = t[1];
        }
    }

    // ---- (img*0.5+0.5)*255, RGB -> YCbCr, round, center at 0 ----
#pragma unroll
    for (int i = 0; i < 8; ++i) {
        float r = ch[0][i] * 127.5f + 127.5f;
        float g = ch[1][i] * 127.5f + 127.5f;
        float b = ch[2][i] * 127.5f + 127.5f;
        float y  = rintf(0.256788f * r + 0.504129f * g + 0.0979059f * b + 16.f);
        float cb = rintf(128.f - 0.148224f * r - 0.290992f * g + 0.439216f * b);
        float cr = rintf(128.f + 0.439216f * r - 0.367788f * g - 0.0714275f * b);
        ch[0][i] = y - 128.f;
        ch[1][i] = cb - 128.f;
        ch[2][i] = cr - 128.f;
    }

    float* L = &lds[wid][0];
    v8f zf[3];

    for (int c = 0; c < 3; ++c) {
        const float* Q = (c == 0) ? Qy : Qc;

        // M1: U = Xc * DD^T          (A = data from memory, B = const)
        v8f u = {};
#pragma unroll
        for (int q = 0; q < 4; ++q) {
            v2f a;
            a[0] = ch[c][2 * q + 0];
            a[1] = ch[c][2 * q + 1];
            u = wmma4(a, dd[q], u);
        }

        // M2: F = DD * U             (A = const, B = shuffled from C/D)
        v8f f = {};
        f = wmma4(dd[0], cd_to_b<0>(u, lane), f);
        f = wmma4(dd[1], cd_to_b<1>(u, lane), f);
        f = wmma4(dd[2], cd_to_b<2>(u, lane), f);
        f = wmma4(dd[3], cd_to_b<3>(u, lane), f);

        // Quantize: F[M][N] -> freq (M&7, N&7) => Q[r][lane&7]
#pragma unroll
        for (int r = 0; r < 8; ++r) {
            float qq = Q[r * 8 + (lane & 7)];
            f[r] = rintf(f[r] / qq) * qq;
        }

        // LDS round-trip: C/D layout -> A layout (in-wave, DS ops are in-order)
#pragma unroll
        for (int r = 0; r < 8; ++r)
            L[(r + rowOff) * LPAD + colN] = f[r];
        asm volatile("s_wait_dscnt 0" ::: "memory");
        v2f g4[4];
#pragma unroll
        for (int q = 0; q < 4; ++q)
            g4[q] = *reinterpret_cast<const v2f*>(&L[m * LPAD + 4 * q + koff]);

        // M3: V = Fq * DD            (A = data from LDS, B = const)
        v8f v = {};
#pragma unroll
        for (int q = 0; q < 4; ++q) v = wmma4(g4[q], ddt[q], v);

        // M4: Z = DD^T * V = idct2   (A = const, B = shuffled)
        v8f z = {};
        z = wmma4(ddt[0], cd_to_b<0>(v, lane), z);
        z = wmma4(ddt[1], cd_to_b<1>(v, lane), z);
        z = wmma4(ddt[2], cd_to_b<2>(v, lane), z);
        z = wmma4(ddt[3], cd_to_b<3>(v, lane), z);
        zf[c] = z;
    }

    // ---- +128, YCbCr -> RGB, round, rescale, store (C/D layout) ----
    const size_t obase =
        (size_t)bi * 3 * plane + (size_t)(ty * 16) * 1024 + (size_t)tx * 16 + colN;
#pragma unroll
    for (int r = 0; r < 8; ++r) {
        float y  = zf[0][r] + 112.f;  // (Z+128) - 16
        float cb = zf[1][r];          // (Z+128) - 128
        float cr = zf[2][r];
        float rr = 1.16438f * y + 3.01124e-07f * cb + 1.59603f * cr;
        float gg = 1.16438f * y - 0.391763f * cb - 0.812968f * cr;
        float bb = 1.16438f * y + 2.01723f * cb + 3.05426e-06f * cr;
        size_t rowi = (size_t)(r + rowOff) * 1024;
        out[obase + 0 * plane + rowi] = rintf(rr) * (2.f / 255.f) - 1.f;
        out[obase + 1 * plane + rowi] = rintf(gg) * (2.f / 255.f) - 1.f;
        out[obase + 2 * plane + rowi] = rintf(bb) * (2.f / 255.f) - 1.f;
    }
}

extern "C" void kernel_launch(void* const* d_in, const int* in_sizes, int n_in,
                              void* d_out, int out_size, void* d_ws, size_t ws_size,
                              hipStream_t stream) {
    (void)n_in; (void)out_size; (void)d_ws; (void)ws_size;
    const float* img = (const float*)d_in[0];
    const float* Qy  = (const float*)d_in[1];
    const float* Qc  = (const float*)d_in[2];
    float* out = (float*)d_out;

    int nImgs = in_sizes[0] / (3 * 1024 * 1024);  // 8 for the reference setup
    int tiles  = nImgs * 64 * 64;                 // one 16x16 tile-triple per wave
    int blocks = (tiles + 7) / 8;                 // 8 waves (256 threads) per block

    hipLaunchKernelGGL(jpeg_wmma_kernel, dim3(blocks), dim3(256), 0, stream,
                       img, Qy, Qc, out, nImgs);
}